// NeighborCooccurrenceEncoder_41308995453071
// MI455X (gfx1250) — compile-verified
//
#include <hip/hip_runtime.h>

// NeighborCooccurrenceEncoder for MI455X (gfx1250, wave32).
//
// Shapes: B=512 rows, S=256 neighbors, D=64 features.
// Phase 1 (per block = one batch row): co-occurrence counts via LDS broadcast.
// Phase 2: per-row GEMM G(256x64) @ w2(64x64) using V_WMMA_F32_16X16X4_F32
//          (fp32 matrix pipe, exact precision; kernel is HBM-write bound at
//          67 MB / 23.3 TB/s ~ 3us so K=4 fp32 WMMA granularity is free).

typedef __attribute__((ext_vector_type(2))) float v2f;
typedef __attribute__((ext_vector_type(8))) float v8f;

#define B_DIM 512
#define S_DIM 256
#define D_DIM 64

__launch_bounds__(256, 2)
__global__ void ncoe_kernel(const int* __restrict__ src_ids,
                            const int* __restrict__ dst_ids,
                            const float* __restrict__ w1,
                            const float* __restrict__ b1,
                            const float* __restrict__ w2,
                            const float* __restrict__ b2,
                            float* __restrict__ out)
{
    __shared__ int   s_src[S_DIM];
    __shared__ int   s_dst[S_DIM];
    __shared__ float s_app_src[S_DIM * 2];   // (s, {in_src, in_dst})
    __shared__ float s_app_dst[S_DIM * 2];
    __shared__ float s_w1[D_DIM];
    __shared__ float s_b1[D_DIM];
    __shared__ float s_b2[D_DIM];
    __shared__ float s_w2[D_DIM * D_DIM];    // row-major (d_in, d_out)

    const int tid = threadIdx.x;      // 0..255
    const int b   = blockIdx.x;       // batch row

    // ---- Stage ids + MLP params into LDS -------------------------------
    s_src[tid] = src_ids[b * S_DIM + tid];
    s_dst[tid] = dst_ids[b * S_DIM + tid];
    if (tid < D_DIM) {
        s_w1[tid] = w1[tid];
        s_b1[tid] = b1[tid];
        s_b2[tid] = b2[tid];
    }
    #pragma unroll
    for (int i = tid; i < D_DIM * D_DIM; i += 256) s_w2[i] = w2[i];
    __syncthreads();

    // ---- Phase 1: co-occurrence counts (all lanes read same LDS word ->
    //      bank broadcast; 256 iters x 4 compares per thread) -------------
    {
        const int sid = s_src[tid];
        const int did = s_dst[tid];
        int css = 0, csd = 0, cds = 0, cdd = 0;
        #pragma unroll 8
        for (int j = 0; j < S_DIM; ++j) {
            const int sj = s_src[j];
            const int dj = s_dst[j];
            css += (sid == sj);
            csd += (sid == dj);
            cds += (did == sj);
            cdd += (did == dj);
        }
        // reference zeroes appearances where id == 0
        s_app_src[tid * 2 + 0] = (sid != 0) ? (float)css : 0.0f;
        s_app_src[tid * 2 + 1] = (sid != 0) ? (float)csd : 0.0f;
        s_app_dst[tid * 2 + 0] = (did != 0) ? (float)cds : 0.0f;
        s_app_dst[tid * 2 + 1] = (did != 0) ? (float)cdd : 0.0f;
    }
    __syncthreads();

    // ---- Phase 2: Out = G @ w2 + 2*b2 with fp32 WMMA -------------------
    // G[s,d] = relu(a0*w1[d]+b1[d]) + relu(a1*w1[d]+b1[d])  (built in regs)
    //
    // V_WMMA_F32_16X16X4_F32 layouts (ISA 7.12.2):
    //   A 16x4 : lane M = lane&15 ; K = 2*(lane>>4) + {vgpr0, vgpr1}
    //   B 4x16 : lane N = lane&15 ; K = 2*(lane>>4) + {vgpr0, vgpr1} (mirrors A)
    //   C/D    : lane N = lane&15 ; rows M = 8*(lane>>4) + vgpr(0..7)
    const int lane = tid & 31;
    const int wave = tid >> 5;        // 0..7
    const int half = lane >> 4;       // 0/1
    const int lcol = lane & 15;

    // 128 tiles: side(2) x mtile(16) x ntile(4); uniform per-wave loop so
    // EXEC stays all-ones around WMMA.
    for (int t = wave; t < 128; t += 8) {
        const int side = t >> 6;      // 0 = src_feat, 1 = dst_feat
        const int rem  = t & 63;
        const int mt   = rem >> 2;    // 0..15 (rows of S)
        const int nt   = rem & 3;     // 0..3  (cols of D)

        const float* APP = side ? s_app_dst : s_app_src;
        const int row = mt * 16 + lcol;
        const float a0 = APP[row * 2 + 0];
        const float a1 = APP[row * 2 + 1];

        v8f acc = {0.f, 0.f, 0.f, 0.f, 0.f, 0.f, 0.f, 0.f};
        #pragma unroll
        for (int k = 0; k < 16; ++k) {             // K = 64 in steps of 4
            const int kv0 = k * 4 + half * 2;
            const int kv1 = kv0 + 1;
            v2f A;
            A.x = fmaxf(fmaf(a0, s_w1[kv0], s_b1[kv0]), 0.0f)
                + fmaxf(fmaf(a1, s_w1[kv0], s_b1[kv0]), 0.0f);
            A.y = fmaxf(fmaf(a0, s_w1[kv1], s_b1[kv1]), 0.0f)
                + fmaxf(fmaf(a1, s_w1[kv1], s_b1[kv1]), 0.0f);
            v2f Bf;
            Bf.x = s_w2[kv0 * D_DIM + nt * 16 + lcol];
            Bf.y = s_w2[kv1 * D_DIM + nt * 16 + lcol];
            acc = __builtin_amdgcn_wmma_f32_16x16x4_f32(
                      /*neg_a=*/false, A, /*neg_b=*/false, Bf,
                      /*c_mod=*/(short)0, acc,
                      /*reuse_a=*/false, /*reuse_b=*/false);
        }

        const float bias2 = 2.0f * s_b2[nt * 16 + lcol];
        float* op = out
                  + (((size_t)side * B_DIM + (size_t)b) * S_DIM
                     + (size_t)(mt * 16 + half * 8)) * D_DIM
                  + nt * 16 + lcol;
        #pragma unroll
        for (int r = 0; r < 8; ++r) {
            op[(size_t)r * D_DIM] = acc[r] + bias2;  // 16-lane / 64B coalesced
        }
    }
}

extern "C" void kernel_launch(void* const* d_in, const int* in_sizes, int n_in,
                              void* d_out, int out_size, void* d_ws, size_t ws_size,
                              hipStream_t stream) {
    const int*   src = (const int*)d_in[0];
    const int*   dst = (const int*)d_in[1];
    const float* w1  = (const float*)d_in[2];
    const float* b1  = (const float*)d_in[3];
    const float* w2  = (const float*)d_in[4];
    const float* b2  = (const float*)d_in[5];
    float* out = (float*)d_out;   // src_feat (B,S,64) then dst_feat (B,S,64)

    ncoe_kernel<<<B_DIM, 256, 0, stream>>>(src, dst, w1, b1, w2, b2, out);
}